// ToBEVConvolution_8529805050237
// MI455X (gfx1250) — compile-verified
//
#include <hip/hip_runtime.h>

// ToBEVConvolution for MI455X (gfx1250, wave32, WMMA).
// Memory-bound problem (~100 MB traffic, 0.41 GFLOP) -> keep exact f32 math
// using V_WMMA_F32_16X16X4_F32; group points by height-bin so 16 points
// sharing one 32x64 weight matrix form one WMMA A-tile.

#define XG    256
#define ZG    256
#define COUT  64
#define CIN   32
#define KBINS 16

typedef __attribute__((ext_vector_type(2))) float v2f;
typedef __attribute__((ext_vector_type(8))) float v8f;

// ---------------- pass 0: zero output + bin counters ----------------
__global__ void zero_kernel(float4* __restrict__ out4, int n4, int* __restrict__ counts) {
    int i = blockIdx.x * blockDim.x + threadIdx.x;
    if (i < n4) out4[i] = make_float4(0.f, 0.f, 0.f, 0.f);
    if (blockIdx.x == 0 && threadIdx.x < KBINS) counts[threadIdx.x] = 0;
}

// ---------------- pass 1: histogram by height bin ----------------
__global__ void count_kernel(const int* __restrict__ coords, int n, int* __restrict__ counts) {
    int i = blockIdx.x * blockDim.x + threadIdx.x;
    if (i < n) atomicAdd(&counts[coords[i * 4 + 1]], 1);   // DIM=1, STRIDE=1
}

// ---------------- pass 2: 16-element exclusive scan ----------------
__global__ void scan_kernel(const int* __restrict__ counts, int* __restrict__ offsets,
                            int* __restrict__ cursor) {
    if (threadIdx.x == 0) {
        int s = 0;
        for (int i = 0; i < KBINS; ++i) { offsets[i] = s; cursor[i] = s; s += counts[i]; }
        offsets[KBINS] = s;
    }
}

// ---------------- pass 3: bucket point ids ----------------
__global__ void scatter_kernel(const int* __restrict__ coords, int n,
                               int* __restrict__ cursor, int* __restrict__ order) {
    int i = blockIdx.x * blockDim.x + threadIdx.x;
    if (i < n) {
        int k = coords[i * 4 + 1];
        int p = atomicAdd(&cursor[k], 1);
        order[p] = i;
    }
}

// ---------------- pass 4: WMMA gemm + BEV scatter-add ----------------
// One wave (32 lanes) handles 16 points that all use weight matrix `kernel[k]`.
// A: 16x32 f32 feats rows; B: 32x64 f32 weights (4 tiles of 32x16);
// C: 16x16 f32 accumulators, 8 chained K=4 WMMAs per tile.
__global__ __launch_bounds__(32) void bev_wmma_kernel(
    const float* __restrict__ feats, const int* __restrict__ coords,
    const float* __restrict__ kern, const int* __restrict__ offsets,
    const int* __restrict__ order, float* __restrict__ out) {

    const int k     = blockIdx.y;                         // height bin
    const int start = offsets[k] + (int)blockIdx.x * 16;  // group of 16 points
    const int end   = offsets[k + 1];
    if (start >= end) return;                             // wave-uniform exit

    const int  lane = threadIdx.x;
    const int  m    = lane & 15;
    const bool hi   = lane >= 16;

    __shared__ int lin_s[16];                             // BEV linear index per point

    int point = -1;
    if (start + m < end) point = order[start + m];

    if (!hi) {
        int lin = -1;
        if (point >= 0) {
            const int4 c = ((const int4*)coords)[point];  // x,y,z,b
            lin = (c.w * XG + c.x) * ZG + c.z;
        }
        lin_s[m] = lin;
    }

    // A fragments: 32-bit A 16x4 layout -> lanes 0-15 hold K,K+1 ; lanes 16-31 hold K+2,K+3
    v2f a[8];
#pragma unroll
    for (int kb = 0; kb < 8; ++kb) { a[kb].x = 0.f; a[kb].y = 0.f; }
    if (point >= 0) {
        const float* fr = feats + (size_t)point * CIN + (hi ? 2 : 0);
#pragma unroll
        for (int kb = 0; kb < 8; ++kb) { a[kb].x = fr[4 * kb]; a[kb].y = fr[4 * kb + 1]; }
    }

    __syncthreads();

    // B fragments: 32-bit B 4x16 layout -> lane column N = lane&15;
    // lanes 0-15 hold rows K,K+1 ; lanes 16-31 hold rows K+2,K+3.
    const float* kb_base = kern + (size_t)k * (CIN * COUT) + (hi ? 2 : 0) * COUT + m;

    v8f acc[4];
#pragma unroll
    for (int t = 0; t < 4; ++t)
#pragma unroll
        for (int r = 0; r < 8; ++r) acc[t][r] = 0.f;

#pragma unroll
    for (int kb = 0; kb < 8; ++kb) {
        const float* bp = kb_base + kb * 4 * COUT;
#pragma unroll
        for (int t = 0; t < 4; ++t) {
            v2f b;
            b.x = bp[t * 16];          // row 4kb + (hi?2:0), col 16t+m
            b.y = bp[COUT + t * 16];   // next row
            acc[t] = __builtin_amdgcn_wmma_f32_16x16x4_f32(
                false, a[kb], false, b, (short)0, acc[t], false, false);
        }
    }

    // C/D layout: VGPR r, lane l -> M = r + 8*(l>=16), N = l&15, channel = 16t + N.
    // Scatter with hardware f32 atomics (inline asm, avoids CAS-loop fallback).
#pragma unroll
    for (int t = 0; t < 4; ++t) {
#pragma unroll
        for (int r = 0; r < 8; ++r) {
            const int M   = r + (hi ? 8 : 0);
            const int lin = lin_s[M];
            if (lin >= 0) {
                float* p = out + (size_t)lin * COUT + t * 16 + m;
                float  v = acc[t][r];
                asm volatile("global_atomic_add_f32 %0, %1, off"
                             :: "v"(p), "v"(v) : "memory");
            }
        }
    }
}

extern "C" void kernel_launch(void* const* d_in, const int* in_sizes, int n_in,
                              void* d_out, int out_size, void* d_ws, size_t ws_size,
                              hipStream_t stream) {
    const float* feats  = (const float*)d_in[0];
    const int*   coords = (const int*)d_in[1];
    const float* kern   = (const float*)d_in[2];
    float*       out    = (float*)d_out;
    const int    N      = in_sizes[0] / CIN;

    int* counts  = (int*)d_ws;           // [16]
    int* offsets = counts + KBINS;       // [17]
    int* cursor  = offsets + KBINS + 1;  // [16]
    int* order   = (int*)d_ws + 64;      // [N]

    const int n4 = out_size / 4;
    zero_kernel<<<(n4 + 255) / 256, 256, 0, stream>>>((float4*)d_out, n4, counts);
    count_kernel<<<(N + 255) / 256, 256, 0, stream>>>(coords, N, counts);
    scan_kernel<<<1, 32, 0, stream>>>(counts, offsets, cursor);
    scatter_kernel<<<(N + 255) / 256, 256, 0, stream>>>(coords, N, cursor, order);

    dim3 grid((N + 15) / 16, KBINS);
    bev_wmma_kernel<<<grid, 32, 0, stream>>>(feats, coords, kern, offsets, order, out);
}